// QSP_Func_Fit_8229157339721
// MI455X (gfx1250) — compile-verified
//
#include <hip/hip_runtime.h>
#include <stdint.h>

typedef float v2f __attribute__((ext_vector_type(2)));
typedef float v8f __attribute__((ext_vector_type(8)));

#define PI_D 3.14159265358979323846
#define INV_SQRT2_D 0.70710678118654752440

// ---------------------------------------------------------------------------
// Kernel A: single wave32. Computes f(x_m) at 64 Chebyshev nodes via the exact
// QSP chain in fp64 (column 0 of M suffices: u = M00, v = M10), then does the
// 64x64 DCT matvec with V_WMMA_F32_16X16X4_F32 to produce the 64 Chebyshev
// coefficients of Re(u00) as a polynomial in a.
// ---------------------------------------------------------------------------
__global__ __launch_bounds__(32) void qsp_coeff_kernel(const float* __restrict__ phi,
                                                       float* __restrict__ coef) {
  __shared__ double cs_[64];
  __shared__ double sn_[64];
  __shared__ float  fv[64];
  const int lane = threadIdx.x;  // 0..31, one full wave -> EXEC all ones

  // cos/sin(phi_k/2), cooperatively
  for (int k = lane; k < 64; k += 32) {
    double h = 0.5 * (double)phi[k];
    cs_[k] = cos(h);
    sn_[k] = sin(h);
  }
  __syncthreads();

  // Chain evaluation at nodes x_m = cos((2m+1)*pi/128); s = sin(theta) = sqrt(1-x^2)
  for (int half = 0; half < 2; ++half) {
    int m = lane + 32 * half;
    double th = (double)(2 * m + 1) * (PI_D / 128.0);
    double a = cos(th);
    double s = sin(th);
    // M0 = RZ(phi0) @ H, column 0: u = e^{-i phi0/2}/sqrt2, v = e^{+i phi0/2}/sqrt2
    double ur =  cs_[0] * INV_SQRT2_D, ui = -sn_[0] * INV_SQRT2_D;
    double vr =  cs_[0] * INV_SQRT2_D, vi =  sn_[0] * INV_SQRT2_D;
    for (int k = 1; k < 64; ++k) {
      // W: u' = a*u + i s*v ; v' = i s*u + a*v
      double tur = a * ur - s * vi;
      double tui = a * ui + s * vr;
      double tvr = a * vr - s * ui;
      double tvi = a * vi + s * ur;
      // RZ: u *= e^{-i phi_k/2}, v *= e^{+i phi_k/2}
      double c = cs_[k], d = sn_[k];
      ur = tur * c + tui * d;
      ui = tui * c - tur * d;
      vr = tvr * c - tvi * d;
      vi = tvi * c + tvr * d;
    }
    fv[m] = (float)((ur + vr) * INV_SQRT2_D);  // Re(u00) at node m
  }
  __syncthreads();

  // DCT matvec via WMMA: coef[k] = w_k * sum_m cos(k*(2m+1)*pi/128) * f[m]
  // A (16x4 fp32 layout): lanes 0-15 hold K=0(x),K=1(y); lanes 16-31 hold K=2,K=3.
  // B (4x16): row K replicated across N columns -> every column of D equals the
  // matvec result; extract column N=0 from lane 0 (M=j) and lane 16 (M=8+j).
  const int r  = lane & 15;
  const int hi = lane >> 4;
#pragma unroll
  for (int g = 0; g < 4; ++g) {
    v8f acc = (v8f)0.0f;
    int kidx = 16 * g + r;
    double wk = (kidx == 0) ? (1.0 / 64.0) : (2.0 / 64.0);
#pragma unroll
    for (int t = 0; t < 16; ++t) {
      int m0 = 4 * t + 2 * hi;
      v2f A, B;
      A.x = (float)(wk * cos((double)(kidx * (2 * m0 + 1)) * (PI_D / 128.0)));
      A.y = (float)(wk * cos((double)(kidx * (2 * m0 + 3)) * (PI_D / 128.0)));
      B.x = fv[m0];
      B.y = fv[m0 + 1];
      acc = __builtin_amdgcn_wmma_f32_16x16x4_f32(false, A, false, B,
                                                  (short)0, acc, false, false);
    }
    if (lane == 0) {
#pragma unroll
      for (int j = 0; j < 8; ++j) coef[16 * g + j] = acc[j];
    } else if (lane == 16) {
#pragma unroll
      for (int j = 0; j < 8; ++j) coef[16 * g + 8 + j] = acc[j];
    }
  }
}

// ---------------------------------------------------------------------------
// Kernel B: Clenshaw evaluation of the 64-term Chebyshev series at a_n.
// 2 elements per thread in float2 (targets v_pk_fma_f32). Coefficients read
// through the constant address space so they live in SGPRs via s_load.
// a_n = w_real[n,0,0] -> flat offset 4n; only 4 useful bytes per 16-byte element.
// ---------------------------------------------------------------------------
__global__ __launch_bounds__(256) void qsp_eval_kernel(const float* __restrict__ wr,
                                                       const float* __restrict__ coef,
                                                       float* __restrict__ out,
                                                       int n) {
  typedef const __attribute__((address_space(4))) float cfloat;
  cfloat* cp = (cfloat*)(uintptr_t)coef;
  float c[64];
#pragma unroll
  for (int k = 0; k < 64; ++k) c[k] = cp[k];

  long long i  = (long long)blockIdx.x * blockDim.x + threadIdx.x;
  long long i0 = 2 * i;
  if (i0 + 1 < n) {
    float a0 = wr[8 * i];       // element 2i:   flat 4*(2i)
    float a1 = wr[8 * i + 4];   // element 2i+1: flat 4*(2i+1)
    v2f a  = {a0, a1};
    v2f ta = a + a;
    v2f b1 = {0.f, 0.f}, b2 = {0.f, 0.f};
#pragma unroll
    for (int k = 63; k >= 1; --k) {
      v2f t = ta * b1 - b2 + c[k];
      b2 = b1;
      b1 = t;
    }
    v2f res = a * b1 - b2 + c[0];
    *reinterpret_cast<v2f*>(out + i0) = res;  // 8B-aligned (i0 even)
  } else if (i0 < n) {  // odd-N tail
    float a0 = wr[8 * i];
    float b1 = 0.f, b2 = 0.f;
#pragma unroll
    for (int k = 63; k >= 1; --k) {
      float t = 2.0f * a0 * b1 - b2 + c[k];
      b2 = b1;
      b1 = t;
    }
    out[i0] = a0 * b1 - b2 + c[0];
  }
}

extern "C" void kernel_launch(void* const* d_in, const int* in_sizes, int n_in,
                              void* d_out, int out_size, void* d_ws, size_t ws_size,
                              hipStream_t stream) {
  const float* phi = (const float*)d_in[0];   // [64] fp32
  const float* wr  = (const float*)d_in[1];   // [N,2,2] fp32 (a = wr[4n])
  float* out  = (float*)d_out;                // [N] fp32
  float* coef = (float*)d_ws;                 // 64 fp32 coefficients (scratch)
  int n = out_size;
  if (n <= 0) return;

  qsp_coeff_kernel<<<1, 32, 0, stream>>>(phi, coef);

  int pairs  = (n + 1) / 2;
  int blocks = (pairs + 255) / 256;
  qsp_eval_kernel<<<blocks, 256, 0, stream>>>(wr, coef, out, n);
}